// PolynomialToRoots_47502338294566
// MI455X (gfx1250) — compile-verified
//
#include <hip/hip_runtime.h>
#include <stdint.h>

// Problem constants (match reference)
#define ORDER   32        // polynomial degree (M); M+1 coefficients
#define NPOLY   32768
#define WPB     8         // waves per block (blockDim = 256)
#define ITERS   64        // Aberth-Ehrlich is cubically convergent; 64 is ample margin

// ---------------------------------------------------------------------------
// One wave32 per polynomial, one lane per root.
// Aberth-Ehrlich simultaneous iteration:
//   N_i = p(z_i)/p'(z_i)
//   z_i <- z_i - N_i / (1 - N_i * sum_{j!=i} 1/(z_i - z_j))
// Coefficients are staged global->LDS via GLOBAL_LOAD_ASYNC_TO_LDS (ASYNCcnt),
// then broadcast-read into per-lane registers. The all-pairs interaction term
// reads the per-wave z tile via vectorized ds_load_b128 broadcasts.
// ---------------------------------------------------------------------------
__global__ __launch_bounds__(32 * WPB)
void PolynomialToRoots_aberth_kernel(const float* __restrict__ a,
                                     float* __restrict__ out)
{
    const int lane = threadIdx.x & 31;
    const int wib  = threadIdx.x >> 5;              // wave in block
    const int poly = blockIdx.x * WPB + wib;        // one polynomial per wave

    __shared__ float coef[WPB][ORDER + 1];          // raw coefficients per wave
    __shared__ float zsh_re[WPB][32];
    __shared__ float zsh_im[WPB][32];

    const float* ap = a + poly * (ORDER + 1);
    __builtin_prefetch(ap, 0, 0);                   // global_prefetch_b8

    // ---- async coefficient staging: 33 floats per wave, one per lane -------
    {
        const uint32_t lds0 = (uint32_t)(uintptr_t)(&coef[wib][lane]);
        const unsigned long long g0 =
            (unsigned long long)(uintptr_t)(ap + lane);
        asm volatile("global_load_async_to_lds_b32 %0, %1, off"
                     :: "v"(lds0), "v"(g0) : "memory");
        if (lane == 0) {                            // 33rd element
            const uint32_t lds1 = (uint32_t)(uintptr_t)(&coef[wib][ORDER]);
            const unsigned long long g1 =
                (unsigned long long)(uintptr_t)(ap + ORDER);
            asm volatile("global_load_async_to_lds_b32 %0, %1, off"
                         :: "v"(lds1), "v"(g1) : "memory");
        }
        asm volatile("s_wait_asynccnt 0" ::: "memory");
        // Per-wave LDS region + wave-level ASYNCcnt completion: no block
        // barrier needed before this wave reads its own coef tile.
    }

    // Normalize to monic: p(z) = z^32 + b[0] z^31 + ... + b[31]
    const float inv_a0 = __builtin_amdgcn_rcpf(coef[wib][0]);
    float b[ORDER];
#pragma unroll
    for (int k = 0; k < ORDER; ++k) b[k] = coef[wib][k + 1] * inv_a0;

    // Initial guesses: circle of radius |b_32|^(1/32) (geometric mean of
    // |roots|), clamped, with an angular offset to avoid symmetric stagnation.
    float r = __powf(fabsf(b[ORDER - 1]) + 1e-12f, 1.0f / (float)ORDER);
    r = fminf(fmaxf(r, 0.25f), 4.0f);
    const float theta = 6.28318530717958647f * ((float)lane + 0.3531f) / (float)ORDER;
    float zr = r * __cosf(theta);
    float zi = r * __sinf(theta);

    for (int it = 0; it < ITERS; ++it) {
        // Publish current estimates (disjoint LDS region per wave).
        zsh_re[wib][lane] = zr;
        zsh_im[wib][lane] = zi;
        // Compiler-visible ordering of publish vs. broadcast reads. The WAR
        // on the next iteration's publish is covered by wave32 lockstep +
        // in-order DS per wave (tile is private to this wave).
        __syncthreads();

        // Horner: p (pr,pi) and p' (dr,di) simultaneously; real coefficients.
        float pr = 1.0f, pi = 0.0f;
        float dr = 0.0f, di = 0.0f;
#pragma unroll
        for (int k = 0; k < ORDER; ++k) {
            // d = d*z + p   (uses previous p)
            const float ndr = fmaf(dr, zr, fmaf(-di, zi, pr));
            const float ndi = fmaf(dr, zi, fmaf( di, zr, pi));
            // p = p*z + b[k]
            const float npr = fmaf(pr, zr, fmaf(-pi, zi, b[k]));
            const float npi = fmaf(pr, zi,  pi * zr);
            dr = ndr; di = ndi; pr = npr; pi = npi;
        }

        // Newton ratio N = p / p'
        const float dinv = __builtin_amdgcn_rcpf(fmaf(dr, dr, fmaf(di, di, 1e-30f)));
        const float nr = (pr * dr + pi * di) * dinv;
        const float ni = (pi * dr - pr * di) * dinv;

        // Aberth sum S = sum_j 1/(z_i - z_j). Self term has zero numerator and
        // epsilon denominator -> contributes exactly 0, so no branch needed.
        // (Compiler vectorizes these broadcasts into ds_load_b128.)
        float sr = 0.0f, si = 0.0f;
#pragma unroll
        for (int j = 0; j < 32; ++j) {
            const float xr = zr - zsh_re[wib][j];
            const float xi = zi - zsh_im[wib][j];
            const float inv = __builtin_amdgcn_rcpf(fmaf(xr, xr, fmaf(xi, xi, 1e-30f)));
            sr = fmaf(xr, inv, sr);
            si = fmaf(-xi, inv, si);
        }

        // w = N / (1 - N*S);  z -= w
        const float tr = 1.0f - (nr * sr - ni * si);
        const float ti = -(nr * si + ni * sr);
        const float tinv = __builtin_amdgcn_rcpf(fmaf(tr, tr, fmaf(ti, ti, 1e-30f)));
        zr -= (nr * tr + ni * ti) * tinv;
        zi -= (ni * tr - nr * ti) * tinv;
    }

    // complex64 layout: interleaved (re, im)
    reinterpret_cast<float2*>(out)[poly * ORDER + lane] = make_float2(zr, zi);
}

// ---------------------------------------------------------------------------
extern "C" void kernel_launch(void* const* d_in, const int* in_sizes, int n_in,
                              void* d_out, int out_size, void* d_ws, size_t ws_size,
                              hipStream_t stream)
{
    (void)in_sizes; (void)n_in; (void)d_ws; (void)ws_size; (void)out_size;
    const float* a   = (const float*)d_in[0];
    float*       out = (float*)d_out;     // complex64 roots, interleaved re/im

    const int blocks = NPOLY / WPB;       // 4096 blocks x 256 threads
    PolynomialToRoots_aberth_kernel<<<blocks, 32 * WPB, 0, stream>>>(a, out);
}